// GeometricWarper_52527450030283
// MI455X (gfx1250) — compile-verified
//
#include <hip/hip_runtime.h>

// GeometricWarper for MI455X (gfx1250, wave32).
// Memory-bound scatter workload (~315 MB total traffic => ~13.5 us @ 23.3 TB/s HBM;
// z-buffer + outputs fit in 192 MB L2 so atomic-min RMW stays on-chip).
// The rigid transform R*p + t runs on the matrix pipe via V_WMMA_F32_16X16X4_F32
// (fp32-exact, matches the fp32 reference), 32 pixels per wave with 2 WMMAs.

typedef __attribute__((ext_vector_type(2))) float v2f;
typedef __attribute__((ext_vector_type(8))) float v8f;

#define B_   16
#define H_   480
#define W_   640
#define HW_  (H_ * W_)        // 307200
#define CHW_ (3 * HW_)
#define N_   (B_ * HW_)       // 4915200 (multiple of 32; grids divide exactly)

// ---------------------------------------------------------------- K0: init
__global__ void warp_init_kernel(unsigned long long* __restrict__ zb) {
  int i = blockIdx.x * blockDim.x + threadIdx.x;
  if (i < N_) zb[i] = ~0ull;  // key = +inf depth, no source
}

// ------------------------------------------------------- K1: project+scatter
// One wave32 = 32 source pixels. A(16x4) rows 0-2 and 8-10 hold [R | t]
// (rows 8-10 duplicated so lanes 16-31, which read D rows 8-15 per the ISA
// C/D layout, receive the transform of their own pixel). B(4x16) columns are
// homogeneous points [X,Y,Z,1]; two WMMAs cover pixel groups 0-15 and 16-31.
__global__ void warp_project_kernel(const float* __restrict__ depth,
                                    const float* __restrict__ Ks,
                                    const float* __restrict__ Kd,
                                    const float* __restrict__ T,
                                    unsigned long long* __restrict__ zb) {
  const int lane = threadIdx.x & 31;
  const int wave = (blockIdx.x * blockDim.x + threadIdx.x) >> 5;
  const long base = (long)wave * 32;              // flattened pixel base (B*HW)
  const int b = (int)(base / HW_);                // wave-uniform (32 | HW)
  const int p = (int)(base - (long)b * HW_) + lane;  // this lane's source pixel
  const bool hi = lane >= 16;

  // Source intrinsics (wave-uniform scalar loads)
  const float fx = Ks[b * 9 + 0], fy = Ks[b * 9 + 4];
  const float cx = Ks[b * 9 + 2], cy = Ks[b * 9 + 5];

  // Per-lane unproject
  const float d  = depth[b * HW_ + p];
  const float uf = (float)(p % W_);
  const float vf = (float)(p / W_);
  const float X  = (uf - cx) * d / fx;
  const float Y  = (vf - cy) * d / fy;

  // ---- A matrix 16x4 (ISA layout: lanes 0-15 carry K=0/K=1 in v0/v1,
  //      lanes 16-31 carry K=2/K=3). Rows {0,1,2,8,9,10} = [R row | t].
  v2f Am; Am.x = 0.0f; Am.y = 0.0f;
  const int m = lane & 15;
  if ((m & 7) < 3) {
    const float* Tb = T + b * 16 + 4 * (m & 7) + (hi ? 2 : 0);
    Am.x = Tb[0];            // K = hi?2:0  -> R[m][0] or R[m][2]
    Am.y = Tb[1];            // K = hi?3:1  -> R[m][1] or t[m]
  }

  // ---- B matrices 4x16 (rows 0,2 in v0; rows 1,3 in v1; columns = points)
  const float dlo = __shfl(d, lane & 15);          // Z row for points 0-15
  const float Xhi = __shfl(X, (lane & 15) + 16);   // X row for points 16-31
  const float Yhi = __shfl(Y, (lane & 15) + 16);   // Y row for points 16-31
  v2f B1, B2;
  B1.x = hi ? dlo : X;    B1.y = hi ? 1.0f : Y;    // points base+0 .. base+15
  B2.x = hi ? d   : Xhi;  B2.y = hi ? 1.0f : Yhi;  // points base+16 .. base+31

  v8f C0 = {0.f, 0.f, 0.f, 0.f, 0.f, 0.f, 0.f, 0.f};
  // D = A x B (+0): rows 0-2 (lanes 0-15) / rows 8-10 (lanes 16-31) hold
  // the transformed point of column n = lane&15.
  v8f D1 = __builtin_amdgcn_wmma_f32_16x16x4_f32(false, Am, false, B1,
                                                 (short)0, C0, false, false);
  v8f D2 = __builtin_amdgcn_wmma_f32_16x16x4_f32(false, Am, false, B2,
                                                 (short)0, C0, false, false);

  const float Xd = hi ? D2[0] : D1[0];
  const float Yd = hi ? D2[1] : D1[1];
  const float Zd = hi ? D2[2] : D1[2];

  // Project into destination view
  const float Zs  = fmaxf(Zd, 1e-6f);
  const float fxd = Kd[b * 9 + 0], fyd = Kd[b * 9 + 4];
  const float cxd = Kd[b * 9 + 2], cyd = Kd[b * 9 + 5];
  const float udf = rintf(fxd * Xd / Zs + cxd);   // round-half-even == jnp.round
  const float vdf = rintf(fyd * Yd / Zs + cyd);

  const bool valid = (d > 0.0f) && (Zd > 1e-6f) &&
                     (udf >= 0.0f) && (udf <= (float)(W_ - 1)) &&
                     (vdf >= 0.0f) && (vdf <= (float)(H_ - 1));
  if (valid) {
    const int ud = (int)udf, vd = (int)vdf;
    // key = (depth bits << 32) | src pixel id: positive-float bits are
    // monotonic as uint, so u64 atomic-min implements the z-buffer with a
    // deterministic tie-break and carries the winner's identity for free.
    const unsigned long long key =
        ((unsigned long long)__float_as_uint(Zd) << 32) | (unsigned)p;
    atomicMin(&zb[(long)b * HW_ + vd * W_ + ud], key);
  }
}

// ----------------------------------------------------------- K2: resolve
// Per destination pixel: decode winner, gather its RGB, emit depth + masks.
// Writes every output element (harness poisons d_out).
__global__ void warp_resolve_kernel(const float* __restrict__ image,
                                    const unsigned long long* __restrict__ zb,
                                    float* __restrict__ out) {
  const int i = blockIdx.x * blockDim.x + threadIdx.x;
  if (i >= N_) return;
  const int b   = i / HW_;
  const int pix = i - b * HW_;

  const unsigned long long key = zb[i];
  float r = 0.f, g = 0.f, bl = 0.f, z = 0.f, vm = 0.f;
  if (key != ~0ull) {
    z = __uint_as_float((unsigned)(key >> 32));
    const int src = (int)(key & 0xffffffffu);
    const float* img = image + (long)b * CHW_;
    r  = img[src];
    g  = img[HW_ + src];
    bl = img[2 * HW_ + src];
    vm = 1.0f;
  }

  float* wr   = out;                          // (B,3,H,W)
  float* wd   = out + (long)B_ * CHW_;        // (B,H,W)
  float* vmo  = wd + (long)N_;                // (B,H,W) valid mask
  float* ivmo = vmo + (long)N_;               // (B,H,W) ~valid

  const long o = (long)b * CHW_ + pix;
  wr[o]            = r;
  wr[o + HW_]      = g;
  wr[o + 2 * HW_]  = bl;
  wd[i]   = z;
  vmo[i]  = vm;
  ivmo[i] = 1.0f - vm;
}

// ---------------------------------------------------------------- launcher
extern "C" void kernel_launch(void* const* d_in, const int* in_sizes, int n_in,
                              void* d_out, int out_size, void* d_ws, size_t ws_size,
                              hipStream_t stream) {
  const float* image = (const float*)d_in[0];   // (16,3,480,640)
  const float* depth = (const float*)d_in[1];   // (16,1,480,640)
  const float* Ks    = (const float*)d_in[2];   // (16,3,3)
  const float* Kd    = (const float*)d_in[3];   // (16,3,3)
  const float* T     = (const float*)d_in[4];   // (16,4,4)

  unsigned long long* zb = (unsigned long long*)d_ws;  // 39.3 MB z-buffer keys

  const int threads = 256;                       // 8 waves per WGP dispatch
  warp_init_kernel<<<(N_ + threads - 1) / threads, threads, 0, stream>>>(zb);
  // One wave per 32 pixels -> N_ threads exactly (N_ % 256 == 0)
  warp_project_kernel<<<N_ / threads, threads, 0, stream>>>(depth, Ks, Kd, T, zb);
  warp_resolve_kernel<<<(N_ + threads - 1) / threads, threads, 0, stream>>>(
      image, zb, (float*)d_out);
}